// GRUHM_20873541059180
// MI455X (gfx1250) — compile-verified
//
#include <hip/hip_runtime.h>

// Problem dims (fixed by the reference).
#define DD 512
#define TT 2048
#define BB 32

typedef __attribute__((ext_vector_type(2))) float v2f;
typedef __attribute__((ext_vector_type(8))) float v8f;

#define LOG2E 1.4426950408889634f

// gfx1250 has hardware v_tanh_f32 (tanh-insts). sigmoid(a) = 0.5*tanh(a/2)+0.5
// — mathematically identical to 1/(1+exp(-a)) but 1 trans op, zero divides.
__device__ __forceinline__ float sigmoid_fast(float a) {
    return __builtin_fmaf(0.5f, __builtin_amdgcn_tanhf(0.5f * a), 0.5f);
}
__device__ __forceinline__ float tanh_fast(float a) {
    return __builtin_amdgcn_tanhf(a);
}
// exp(-relu(u)) = exp2(-log2(e) * max(u,0)) -> one v_exp_f32.
__device__ __forceinline__ float exp_neg_relu(float u) {
    return __builtin_amdgcn_exp2f(-LOG2E * fmaxf(u, 0.0f));
}

struct GrudParams {
    const float* x;       // [B,3,D,T]
    const float* w_dg_x; const float* w_dg_h;
    const float* w_xz;   const float* w_hz;   const float* w_mz;
    const float* w_xr;   const float* w_hr;   const float* w_mr;
    const float* w_xh;   const float* w_hh;   const float* w_mh;
    const float* b_dg_x; const float* b_dg_h;
    const float* b_z;    const float* b_r;    const float* b_h;
    const float* x_mean;
    float* h_out;         // [B,D] final hidden state
};

// One thread per (b,d) channel: fully independent scalar GRU-D recurrence
// (all weights are diagonal). Each thread streams its contiguous T-row of
// X/M/Delta with float4 loads; a 128B line serves 8 iterations of one thread
// and stays hot in WGP$; explicit prefetch covers the serial walk.
__global__ __launch_bounds__(256) void grud_scan_kernel(GrudParams p) {
    const int tid = blockIdx.x * blockDim.x + threadIdx.x;   // 0..B*D-1
    const int d = tid & (DD - 1);
    const int b = tid >> 9;                                  // /DD

    const float wdgx = p.w_dg_x[d], wdgh = p.w_dg_h[d];
    const float wxz = p.w_xz[d], whz = p.w_hz[d], wmz = p.w_mz[d];
    const float wxr = p.w_xr[d], whr = p.w_hr[d], wmr = p.w_mr[d];
    const float wxh = p.w_xh[d], whh = p.w_hh[d], wmh = p.w_mh[d];
    const float bdgx = p.b_dg_x[d], bdgh = p.b_dg_h[d];
    const float bz = p.b_z[d], br = p.b_r[d], bh = p.b_h[d];
    const float xmean = p.x_mean[d];

    const size_t rowX = ((size_t)(b * 3 + 0) * DD + d) * TT;
    const size_t rowM = ((size_t)(b * 3 + 1) * DD + d) * TT;
    const size_t rowD = ((size_t)(b * 3 + 2) * DD + d) * TT;
    const float4* Xp = (const float4*)(p.x + rowX);
    const float4* Mp = (const float4*)(p.x + rowM);
    const float4* Dp = (const float4*)(p.x + rowD);

    float h = 0.0f;
    float xl = 0.0f;

    auto step = [&](float xt, float m, float dlt) {
        float gx = exp_neg_relu(__builtin_fmaf(wdgx, dlt, bdgx));
        float gh = exp_neg_relu(__builtin_fmaf(wdgh, dlt, bdgh));
        // gx*xl + (1-gx)*xmean == xmean + gx*(xl - xmean)
        float px   = __builtin_fmaf(gx, xl - xmean, xmean);
        // m*xt + (1-m)*px == px + m*(xt - px)
        float xhat = __builtin_fmaf(m, xt - px, px);
        float hd = gh * h;                                   // decayed hidden
        float az = __builtin_fmaf(wxz, xhat,
                   __builtin_fmaf(whz, hd,
                   __builtin_fmaf(wmz, m, bz)));
        float ar = __builtin_fmaf(wxr, xhat,
                   __builtin_fmaf(whr, hd,
                   __builtin_fmaf(wmr, m, br)));
        float z = sigmoid_fast(az);
        float r = sigmoid_fast(ar);
        float ah = __builtin_fmaf(wxh, xhat,
                   __builtin_fmaf(whh, r * hd,
                   __builtin_fmaf(wmh, m, bh)));
        float ht = tanh_fast(ah);
        // (1-z)*hd + z*ht == hd + z*(ht - hd)
        h  = __builtin_fmaf(z, ht - hd, hd);
        xl = __builtin_fmaf(m, xt - xl, xl);
    };

    for (int i = 0; i < TT / 4; ++i) {
        if ((i & 7) == 0) {
            __builtin_prefetch(Xp + i + 8, 0, 0);
            __builtin_prefetch(Mp + i + 8, 0, 0);
            __builtin_prefetch(Dp + i + 8, 0, 0);
        }
        float4 xv = Xp[i];
        float4 mv = Mp[i];
        float4 dv = Dp[i];
        step(xv.x, mv.x, dv.x);
        step(xv.y, mv.y, dv.y);
        step(xv.z, mv.z, dv.z);
        step(xv.w, mv.w, dv.w);
    }

    p.h_out[tid] = h;
}

// Final projection out[b] = sum_d H[b,d]*w[d] + bias via V_WMMA_F32_16X16X4_F32
// (full fp32, exact same precision class as the reference matvec).
// Two waves; wave w owns batch rows [16w, 16w+16). K accumulated 4 at a time.
// A (16x4 f32): lanes 0-15 carry {K=0,K=1} for M=lane, lanes 16-31 carry
// {K=2,K=3} for M=lane-16. B (4x16) mirrors this per-lane over N; only column
// N=0 (lanes 0 and 16) carries lin3_w, other columns are zero.
// C/D (16x16 f32): VGPR v of lane n<16 holds D[M=v,N=n]; lane n>=16 holds
// D[M=v+8,N=n-16] — so result column 0 lives in lanes 0 and 16.
__global__ __launch_bounds__(64) void grud_proj_wmma_kernel(
        const float* __restrict__ H,      // [32,512]
        const float* __restrict__ w,      // [512]
        const float* __restrict__ bias,   // [1]
        float* __restrict__ out) {        // [32]
    const int lane = threadIdx.x & 31;
    const int wave = threadIdx.x >> 5;
    const int row0 = wave * 16;
    const int m = lane & 15;
    const int khalf = (lane >> 4) << 1;              // 0 for lanes 0-15, 2 for 16-31
    const float wsel = (m == 0) ? 1.0f : 0.0f;       // only column N==0 of B is live
    const float* hrow = H + (size_t)(row0 + m) * DD;

    v8f acc = {0.f, 0.f, 0.f, 0.f, 0.f, 0.f, 0.f, 0.f};
    for (int k0 = 0; k0 < DD; k0 += 4) {
        const int k = k0 + khalf;
        v2f a; a.x = hrow[k];       a.y = hrow[k + 1];
        v2f bv; bv.x = w[k] * wsel; bv.y = w[k + 1] * wsel;
        acc = __builtin_amdgcn_wmma_f32_16x16x4_f32(
            /*neg_a=*/false, a, /*neg_b=*/false, bv,
            /*c_mod=*/(short)0, acc, /*reuse_a=*/false, /*reuse_b=*/false);
    }

    const float bb = bias[0];
    if (lane == 0) {
#pragma unroll
        for (int i = 0; i < 8; ++i) out[row0 + i] = acc[i] + bb;        // M=0..7
    } else if (lane == 16) {
#pragma unroll
        for (int i = 0; i < 8; ++i) out[row0 + 8 + i] = acc[i] + bb;    // M=8..15
    }
}

extern "C" void kernel_launch(void* const* d_in, const int* in_sizes, int n_in,
                              void* d_out, int out_size, void* d_ws, size_t ws_size,
                              hipStream_t stream) {
    GrudParams p;
    p.x      = (const float*)d_in[0];
    p.w_dg_x = (const float*)d_in[1];
    p.w_dg_h = (const float*)d_in[2];
    p.w_xz   = (const float*)d_in[3];
    p.w_hz   = (const float*)d_in[4];
    p.w_mz   = (const float*)d_in[5];
    p.w_xr   = (const float*)d_in[6];
    p.w_hr   = (const float*)d_in[7];
    p.w_mr   = (const float*)d_in[8];
    p.w_xh   = (const float*)d_in[9];
    p.w_hh   = (const float*)d_in[10];
    p.w_mh   = (const float*)d_in[11];
    p.b_dg_x = (const float*)d_in[12];
    p.b_dg_h = (const float*)d_in[13];
    p.b_z    = (const float*)d_in[14];
    p.b_r    = (const float*)d_in[15];
    p.b_h    = (const float*)d_in[16];
    p.x_mean = (const float*)d_in[17];
    const float* lin3_w = (const float*)d_in[18];
    const float* lin3_b = (const float*)d_in[19];

    p.h_out = (float*)d_ws;   // [B,D] fp32 = 64 KB scratch

    grud_scan_kernel<<<(BB * DD) / 256, 256, 0, stream>>>(p);
    grud_proj_wmma_kernel<<<1, 64, 0, stream>>>((const float*)d_ws, lin3_w, lin3_b,
                                                (float*)d_out);
}